// OnsetVelocityNmsDecoder_28741921144951
// MI455X (gfx1250) — compile-verified
//
#include <hip/hip_runtime.h>
#include <stdint.h>

// Problem constants (match reference).
#define BB 8
#define KK 88
#define TT 32768
#define NROWS (BB * KK)           // 704
#define NTOT (NROWS * TT)         // 23,068,672

#define TILE 2048
#define HALO 4
#define REG (TILE + 2 * HALO)     // 2056 floats per staged array
#define THREADS 256

// Gaussian weights: ksize=7, std=1, normalized (precomputed in double, stored f32).
#define W0 0.0044330482f
#define W1 0.0540055800f
#define W2 0.2420362300f
#define W3 0.3990502800f

// ---------------- CDNA5 async global -> LDS helpers (inline asm, gfx1250) ---

// Default (RT) temporal hint: used for onset, which pass 1 leaves resident in
// the 192MB L2 (92MB tensor) so these should hit L2.
__device__ __forceinline__ void async_ld_b128(uint32_t lds_off, const float* g) {
    asm volatile("global_load_async_to_lds_b128 %0, %1, off"
                 :: "v"(lds_off), "v"((uint64_t)(uintptr_t)g)
                 : "memory");
}

__device__ __forceinline__ void async_ld_b32(uint32_t lds_off, const float* g) {
    asm volatile("global_load_async_to_lds_b32 %0, %1, off"
                 :: "v"(lds_off), "v"((uint64_t)(uintptr_t)g)
                 : "memory");
}

// Non-temporal variants: used for velmap, which is read exactly once — don't
// let it evict the L2-resident onset tensor.
__device__ __forceinline__ void async_ld_b128_nt(uint32_t lds_off, const float* g) {
    asm volatile("global_load_async_to_lds_b128 %0, %1, off th:TH_LOAD_NT"
                 :: "v"(lds_off), "v"((uint64_t)(uintptr_t)g)
                 : "memory");
}

__device__ __forceinline__ void async_ld_b32_nt(uint32_t lds_off, const float* g) {
    asm volatile("global_load_async_to_lds_b32 %0, %1, off th:TH_LOAD_NT"
                 :: "v"(lds_off), "v"((uint64_t)(uintptr_t)g)
                 : "memory");
}

__device__ __forceinline__ void wait_asynccnt0() {
    asm volatile("s_wait_asynccnt 0" ::: "memory");
}

__device__ __forceinline__ uint32_t lds_off_of(const void* p) {
    // Generic pointer to __shared__ = {SHARED_BASE hi32, lds_offset lo32};
    // low 32 bits are the byte offset DS/async instructions want.
    return (uint32_t)(uintptr_t)p;
}

// ---------------- Pass 1: flag = (max(onset) > 0) ---------------------------
// For finite inputs, max(x) > 0  <=>  any(x > 0); reduce that boolean.
// Side benefit: streams the whole onset tensor through L2 with RT policy, so
// pass 2's onset reads hit L2 (onset = 92MB < 192MB global L2).

__global__ void onset_flag_kernel(const float* __restrict__ x, unsigned* __restrict__ flag, int n4) {
    __shared__ int s_any;
    if (threadIdx.x == 0) s_any = 0;
    __syncthreads();

    const float4* __restrict__ x4 = (const float4*)x;
    bool any = false;
    int stride = gridDim.x * blockDim.x;
    for (int i = blockIdx.x * blockDim.x + threadIdx.x; i < n4; i += stride) {
        float4 v = x4[i];
        any |= (v.x > 0.0f) || (v.y > 0.0f) || (v.z > 0.0f) || (v.w > 0.0f);
    }
    unsigned long long b = __ballot(any);           // wave32: low 32 bits
    if (((threadIdx.x & 31) == 0) && b != 0ull) s_any = 1;
    __syncthreads();
    if (threadIdx.x == 0 && s_any) atomicOr(flag, 1u);
}

// ---------------- Pass 2: fused blur + NMS + windowed velocity --------------

__global__ void __launch_bounds__(THREADS)
onset_nms_kernel(const float* __restrict__ onset,
                 const float* __restrict__ vel,
                 const float* __restrict__ pthresh_p,
                 const unsigned* __restrict__ flag_p,
                 float* __restrict__ probs_out,
                 float* __restrict__ vels_out) {
    __shared__ float s_on[REG];
    __shared__ float s_ve[REG];

    const int row = blockIdx.y;                 // 0..NROWS-1
    const int tile0 = blockIdx.x * TILE;        // tile start along T
    const float* __restrict__ on_row = onset + (size_t)row * TT;
    const float* __restrict__ ve_row = vel + (size_t)row * TT;

    // ---- stage tile + 4-halo of both arrays into LDS via async DMA ----
    const bool interior = (tile0 - HALO >= 0) && (tile0 + TILE + HALO <= TT);
    if (interior) {
        // 2056/4 = 514 float4 transfers per array, 16B aligned.
        for (int i = threadIdx.x; i < REG / 4; i += THREADS) {
            int g = tile0 - HALO + i * 4;
            async_ld_b128(lds_off_of(&s_on[i * 4]), on_row + g);
            async_ld_b128_nt(lds_off_of(&s_ve[i * 4]), ve_row + g);
        }
    } else {
        // Row-edge tiles: per-element with reflect indexing (pad mode="reflect").
        for (int i = threadIdx.x; i < REG; i += THREADS) {
            int g = tile0 - HALO + i;
            int gr = (g < 0) ? -g : ((g >= TT) ? (2 * TT - 2 - g) : g);
            async_ld_b32(lds_off_of(&s_on[i]), on_row + gr);
            async_ld_b32_nt(lds_off_of(&s_ve[i]), ve_row + gr);
        }
    }
    wait_asynccnt0();       // this wave's ASYNCcnt == 0 -> its LDS writes landed
    __syncthreads();        // all waves' data visible

    const float th = pthresh_p[0];
    const int useBlur = (int)(flag_p[0]);       // uniform scalar
    const float NEG_INF = -__builtin_huge_valf();

    // lp = e*256 + tid: consecutive lanes hit consecutive LDS banks (no conflicts)
    // and stores are perfectly coalesced.
    for (int e = 0; e < TILE / THREADS; ++e) {
        const int lp = e * THREADS + threadIdx.x;   // 0..TILE-1
        const int t = tile0 + lp;

        // 9-sample onset window centered at s_on[lp+4] (global t).
        float x0 = s_on[lp + 0], x1 = s_on[lp + 1], x2 = s_on[lp + 2];
        float x3 = s_on[lp + 3], x4 = s_on[lp + 4], x5 = s_on[lp + 5];
        float x6 = s_on[lp + 6], x7 = s_on[lp + 7], x8 = s_on[lp + 8];

        // blur at t-1, t, t+1 (reflect padding already baked into the halo)
        float bl = W0 * x0 + W1 * x1 + W2 * x2 + W3 * x3 + W2 * x4 + W1 * x5 + W0 * x6;
        float bc = W0 * x1 + W1 * x2 + W2 * x3 + W3 * x4 + W2 * x5 + W1 * x6 + W0 * x7;
        float br = W0 * x2 + W1 * x3 + W2 * x4 + W3 * x5 + W2 * x6 + W1 * x7 + W0 * x8;

        // probs = where(prev_max > 0, blurred, onset)
        float vl = useBlur ? bl : x3;
        float vc = useBlur ? bc : x4;
        float vr = useBlur ? br : x5;

        // NMS: 3-window max with constant -inf padding at array edges.
        if (t == 0)       vl = NEG_INF;
        if (t == TT - 1)  vr = NEG_INF;
        float p = (vc >= vl && vc >= vr) ? vc : 0.0f;
        p = (p >= th) ? p : 0.0f;

        // windowed velocity mean over [t-1, t+1] with reflect padding.
        float v = (s_ve[lp + 3] + s_ve[lp + 4] + s_ve[lp + 5]) * (1.0f / 3.0f);
        v = (p > 0.0f) ? v : 0.0f;

        const size_t idx = (size_t)row * TT + t;
        // Outputs are write-once / never re-read on device: non-temporal stores
        // keep the L2-resident onset tensor from being evicted mid-pass.
        __builtin_nontemporal_store(p, &probs_out[idx]);
        __builtin_nontemporal_store(v, &vels_out[idx]);
    }
}

// ---------------- launch ----------------------------------------------------

extern "C" void kernel_launch(void* const* d_in, const int* in_sizes, int n_in,
                              void* d_out, int out_size, void* d_ws, size_t ws_size,
                              hipStream_t stream) {
    const float* onset = (const float*)d_in[0];
    const float* vel = (const float*)d_in[1];
    const float* pthresh = (const float*)d_in[2];
    unsigned* flag = (unsigned*)d_ws;

    float* probs_out = (float*)d_out;
    float* vels_out = probs_out + (size_t)NTOT;

    // Zero the flag word (graph-capture-safe memset node), then reduce into it.
    hipMemsetAsync(flag, 0, sizeof(unsigned), stream);

    const int n4 = NTOT / 4;
    onset_flag_kernel<<<2048, THREADS, 0, stream>>>(onset, flag, n4);

    dim3 grid(TT / TILE, NROWS);    // 16 x 704
    onset_nms_kernel<<<grid, THREADS, 0, stream>>>(onset, vel, pthresh, flag,
                                                   probs_out, vels_out);
}